// RandomHingeFern_69114613728657
// MI455X (gfx1250) — compile-verified
//
#include <hip/hip_runtime.h>
#include <hip/hip_bf16.h>
#include <stdint.h>

// Problem constants from the reference
#define B_     4096
#define C_IN   512
#define O_     1024
#define D_     10
#define L_     1024   // 2^D leaves
#define BROWS  8      // batch rows staged per block (16 KB LDS, contiguous in x)
#define OCHNK  2      // o-chunks of 256 per block (o-dim split across 2 blocks)

__global__ __launch_bounds__(256) void RandomHingeFern_kernel(
    const float* __restrict__ x,          // (B, C_IN)
    const float* __restrict__ thresholds, // (O, D)
    const int*   __restrict__ ordinals,   // (O, D)
    const float* __restrict__ weights,    // (O, L)
    float*       __restrict__ out)        // (B, O)
{
    __shared__ float lds_x[BROWS * C_IN];   // 16 KB

    const int tid   = threadIdx.x;
    const int obase = blockIdx.x * (OCHNK * 256);   // 0 or 512
    const int b0    = blockIdx.y * BROWS;

    // ---- Stage x[b0 .. b0+7, :] (16 KB, contiguous) into LDS via CDNA5 async DMA ----
    // 256 threads * 16 B * 4 issues = 16 KB. EXEC all-1s, no divergence yet.
    // INST_OFFSET is added to BOTH global and LDS addresses (ISA 15.18).
    {
        uint64_t g0 = (uint64_t)(uintptr_t)(x + (size_t)b0 * C_IN) + (uint32_t)tid * 16u;
        uint32_t l0 = (uint32_t)(uintptr_t)lds_x + (uint32_t)tid * 16u;
        asm volatile("global_load_async_to_lds_b128 %0, %1, off"
                     :: "v"(l0), "v"(g0) : "memory");
        asm volatile("global_load_async_to_lds_b128 %0, %1, off offset:4096"
                     :: "v"(l0), "v"(g0) : "memory");
        asm volatile("global_load_async_to_lds_b128 %0, %1, off offset:8192"
                     :: "v"(l0), "v"(g0) : "memory");
        asm volatile("global_load_async_to_lds_b128 %0, %1, off offset:12288"
                     :: "v"(l0), "v"(g0) : "memory");
    }

    // ---- Overlap with the DMA: load thr/ord for both o-chunks into registers ----
    // (read-only, independent of LDS, so legal to issue before the barrier)
    float thr[OCHNK][D_];
    int   ord[OCHNK][D_];
    #pragma unroll
    for (int t = 0; t < OCHNK; ++t) {
        const int o = obase + tid + t * 256;
        const float* thrp = thresholds + (size_t)o * D_;
        const int*   ordp = ordinals   + (size_t)o * D_;
        #pragma unroll
        for (int d = 0; d < D_; ++d) { thr[t][d] = thrp[d]; ord[t][d] = ordp[d]; }
    }

    asm volatile("s_wait_asynccnt 0" ::: "memory");
    __syncthreads();

    // ---- Per o-chunk: reuse thr/ord registers across BROWS batch rows ----
    #pragma unroll
    for (int t = 0; t < OCHNK; ++t) {
        const int o = obase + tid + t * 256;
        const float* wrow = weights + (size_t)o * L_;

        #pragma unroll
        for (int bi = 0; bi < BROWS; ++bi) {
            float best = __builtin_inff();
            int   leaf = 0;

            #pragma unroll
            for (int d = 0; d < D_; ++d) {
                // ds gather: VGPR addr = ord*4 + base; bi*2048 folds into DS offset,
                // adjacent bi pair -> ds_load_2addr_stride64_b32
                const float m = lds_x[bi * C_IN + ord[t][d]] - thr[t][d];
                leaf = (leaf << 1) | (m > 0.0f ? 1 : 0);   // d=0 is MSB
                best = fminf(best, fabsf(m));              // |m| folds as src modifier
            }

            const float w = wrow[leaf];                    // L2-resident gather
            // write-once output: non-temporal store, keep L2 for weights/thr/ord
            __builtin_nontemporal_store(w * best, &out[(size_t)(b0 + bi) * O_ + o]);
        }
    }
}

extern "C" void kernel_launch(void* const* d_in, const int* in_sizes, int n_in,
                              void* d_out, int out_size, void* d_ws, size_t ws_size,
                              hipStream_t stream)
{
    const float* x          = (const float*)d_in[0];
    const float* thresholds = (const float*)d_in[1];
    const int*   ordinals   = (const int*)  d_in[2];
    const float* weights    = (const float*)d_in[3];
    float*       out        = (float*)d_out;

    (void)in_sizes; (void)n_in; (void)out_size; (void)d_ws; (void)ws_size;

    dim3 grid(O_ / (OCHNK * 256), B_ / BROWS);   // (2, 512) = 1024 blocks
    RandomHingeFern_kernel<<<grid, dim3(256), 0, stream>>>(
        x, thresholds, ordinals, weights, out);
}